// VectorQuantizer_46969762349475
// MI455X (gfx1250) — compile-verified
//
#include <hip/hip_runtime.h>

typedef __attribute__((ext_vector_type(2))) float v2f;
typedef __attribute__((ext_vector_type(8))) float v8f;

#define K_CODES 512
#define DIM 64
#define HW 4096            // H*W = 64*64
#define ROWS_PER_BLOCK 256
#define THREADS 512        // 16 waves -> 4 per SIMD
#define PAD 68             // LDS row stride (floats): conflict-free WMMA operand reads

__global__ __launch_bounds__(THREADS)
void vq_wmma_f32_kernel(const float* __restrict__ x,
                        const float* __restrict__ emb,
                        float* __restrict__ out,
                        long long half_out) {
    __shared__ float s_x[ROWS_PER_BLOCK * PAD];   // 69632 B: x tile transposed [r][d]
    __shared__ float s_e[K_CODES * PAD];          // 139264 B: full codebook [c][d]
    __shared__ float s_hnorm[K_CODES];            // 0.5*||e_c||^2
    __shared__ int   s_idx[ROWS_PER_BLOCK];       // winning code per row

    const int t  = threadIdx.x;
    const int n0 = blockIdx.x * ROWS_PER_BLOCK;   // flat row base over (b,h,w)
    const int b  = n0 >> 12;                      // / HW
    const int p0 = n0 & (HW - 1);                 // % HW

    // ---- Stage full codebook into LDS (read emb from global exactly once) ----
    for (int j = t; j < K_CODES * DIM; j += THREADS) {
        int c = j >> 6;          // / 64
        int d = j & 63;
        s_e[c * PAD + d] = emb[j];
    }
    // ---- Stage x tile transposed: s_x[r][d] = x[b][d][p0+r] (coalesced) ----
    const float* xb = x + (size_t)b * DIM * HW;
    for (int j = t; j < ROWS_PER_BLOCK * DIM; j += THREADS) {
        int d = j >> 8;          // / 256
        int r = j & 255;
        s_x[r * PAD + d] = xb[(size_t)d * HW + p0 + r];
    }
    __syncthreads();

    // ---- Half squared norms from LDS copy ----
    if (t < K_CODES) {
        const float* er = s_e + t * PAD;
        float s = 0.0f;
        #pragma unroll
        for (int d = 0; d < DIM; ++d) { float v = er[d]; s += v * v; }
        s_hnorm[t] = 0.5f * s;
    }
    __syncthreads();

    // ---- Per-wave 16-row tile: WMMA f32 16x16x4 over all 512 codes ----
    const int lane = t & 31;
    const int wave = t >> 5;          // 0..15 -> rows wave*16 .. +15
    const int n    = lane & 15;       // A row m / B & C column
    const int kk   = lane >> 4;       // K-half selector (ISA A/B VGPR striping)
    const int rowBase = wave * 16;

    float best_val[8];
    int   best_idx[8];
    #pragma unroll
    for (int j = 0; j < 8; ++j) { best_val[j] = -3.402823466e38f; best_idx[j] = 0; }

    // A operand: lane holds x_tile[m=n][k = 4s + 2kk + {0,1}] — hoist to 32 VGPRs
    const float* a_src = s_x + (rowBase + n) * PAD + 2 * kk;
    v2f Av[16];
    #pragma unroll
    for (int s = 0; s < 16; ++s) Av[s] = *(const v2f*)(a_src + 4 * s);

    // B operand base: lane holds e[c0+n][4s + 2kk + {0,1}]
    const float* b_base = s_e + n * PAD + 2 * kk;

    for (int cc = 0; cc < 32; ++cc) {
        const int   code = cc * 16 + n;
        const float* bp  = b_base + cc * 16 * PAD;

        // Two interleaved accumulator chains (halved WMMA RAW depth)
        v8f acc0 = {0.f,0.f,0.f,0.f,0.f,0.f,0.f,0.f};
        v8f acc1 = {0.f,0.f,0.f,0.f,0.f,0.f,0.f,0.f};
        #pragma unroll
        for (int s = 0; s < 8; ++s) {
            v2f b0 = *(const v2f*)(bp + 8 * s);        // K-step 2s   (ds_load_b64)
            v2f b1 = *(const v2f*)(bp + 8 * s + 4);    // K-step 2s+1
            acc0 = __builtin_amdgcn_wmma_f32_16x16x4_f32(
                false, Av[2 * s],     false, b0, (short)0, acc0, false, false);
            acc1 = __builtin_amdgcn_wmma_f32_16x16x4_f32(
                false, Av[2 * s + 1], false, b1, (short)0, acc1, false, false);
        }

        // argmin(dist) == argmax(S - 0.5*||e||^2); column code same for all 8 slots
        const float hn = s_hnorm[code];
        #pragma unroll
        for (int j = 0; j < 8; ++j) {
            float v = (acc0[j] + acc1[j]) - hn;
            if (v > best_val[j]) { best_val[j] = v; best_idx[j] = code; }
        }
    }

    // ---- Butterfly argmax across the 16 lanes striping each row ----
    // C-tile slot j = row (rowBase + j + 8*kk), columns across lanes n=0..15.
    #pragma unroll
    for (int j = 0; j < 8; ++j) {
        float v  = best_val[j];
        int   id = best_idx[j];
        #pragma unroll
        for (int m = 1; m < 16; m <<= 1) {
            float ov = __shfl_xor(v, m, 32);
            int   oi = __shfl_xor(id, m, 32);
            // tie-break to lowest index (jnp.argmin first-occurrence)
            if (ov > v || (ov == v && oi < id)) { v = ov; id = oi; }
        }
        if (n == 0) s_idx[rowBase + j + 8 * kk] = id;
    }
    __syncthreads();

    // ---- Gather winning codebook rows from LDS, write both outputs coalesced ----
    float* out2 = out + half_out;
    for (int j = t; j < ROWS_PER_BLOCK * DIM; j += THREADS) {
        int d = j >> 8;
        int r = j & 255;
        float v = s_e[s_idx[r] * PAD + d];
        size_t o = (size_t)(b * DIM + d) * HW + p0 + r;
        out[o]  = v;
        out2[o] = v;
    }
}

extern "C" void kernel_launch(void* const* d_in, const int* in_sizes, int n_in,
                              void* d_out, int out_size, void* d_ws, size_t ws_size,
                              hipStream_t stream) {
    const float* x   = (const float*)d_in[0];   // [32, 64, 64, 64] fp32
    const float* emb = (const float*)d_in[1];   // [512, 64] fp32

    const int x_elems = in_sizes[0];            // 8388608
    const int n_rows  = x_elems / DIM;          // 131072
    const int blocks  = n_rows / ROWS_PER_BLOCK;// 512
    const long long half_out = (long long)x_elems;

    vq_wmma_f32_kernel<<<dim3(blocks), dim3(THREADS), 0, stream>>>(
        x, emb, (float*)d_out, half_out);
}